// GCN_61967788147120
// MI455X (gfx1250) — compile-verified
//
#include <hip/hip_runtime.h>
#include <hip/hip_bf16.h>
#include <math.h>

typedef __attribute__((ext_vector_type(2))) float v2f;
typedef __attribute__((ext_vector_type(8))) float v8f;

// ---------------------------------------------------------------------------
// Dense node-feature transform using V_WMMA_F32_16X16X4_F32.
// Each wave32 transforms 16 nodes: A = 16xK (K<=4) node features,
// B = KxFOUT weight (padded to 4x16), D = 16x16 f32 (only cols < FOUT used).
// A layout: lanes 0-15 hold K=0,1 in v[0],v[1]; lanes 16-31 hold K=2,3.
// B layout: lane n (n=lane%16) holds W[K][n]; same K split by lane half.
// D layout: VGPR r -> row M=r (lanes 0-15) / M=8+r (lanes 16-31), col=lane%16.
// All template params compile-time => branchless (cndmask) loads, full EXEC
// at the WMMA, and shift/mad store addressing.
// ---------------------------------------------------------------------------
template<int SIN, int FIN, int FOUT, int SOUT, int PAD, bool BIAS>
__global__ void gcn_wmma_transform(const float* __restrict__ hin,
                                   const float* __restrict__ W,
                                   const float* __restrict__ bias,
                                   float* __restrict__ out, int N)
{
    const int lane = threadIdx.x & 31;
    const int wave = threadIdx.x >> 5;
    const int base = (blockIdx.x * (blockDim.x >> 5) + wave) * 16;
    if (base >= N) return;                 // wave-uniform exit; EXEC stays full

    const int m  = lane & 15;              // col (B/D) or row (A)
    const int kp = lane >> 4;              // which K pair this half-wave holds
    const int k0 = 2 * kp, k1 = k0 + 1;
    const long node = base + m;

    // clamped-address loads (uniform across wave) + select-to-zero
    const int ka0 = (k0 < FIN) ? k0 : 0;
    const int ka1 = (k1 < FIN) ? k1 : 0;
    v2f a;
    {
        float v0 = hin[node * SIN + ka0];
        float v1 = hin[node * SIN + ka1];
        a.x = (k0 < FIN) ? v0 : 0.0f;
        a.y = (k1 < FIN) ? v1 : 0.0f;
    }
    v2f b;
    {
        const bool p0 = (k0 < FIN) && (m < FOUT);
        const bool p1 = (k1 < FIN) && (m < FOUT);
        float w0 = W[p0 ? (k0 * FOUT + m) : 0];
        float w1 = W[p1 ? (k1 * FOUT + m) : 0];
        b.x = p0 ? w0 : 0.0f;
        b.y = p1 ? w1 : 0.0f;
    }

    v8f c = {};
    // (neg_a, A, neg_b, B, c_mod, C, reuse_a, reuse_b)
    v8f d = __builtin_amdgcn_wmma_f32_16x16x4_f32(
        false, a, false, b, (short)0, c, false, false);

    constexpr int WLIM = (PAD > FOUT) ? PAD : FOUT;
    if (m < WLIM) {
        float bb = 0.0f;
        if constexpr (BIAS) { if (m < FOUT) bb = bias[m]; }
        const int mbase = base + 8 * kp;
#pragma unroll
        for (int r = 0; r < 8; ++r) {
            float val = (m < FOUT) ? (d[r] + bb) : 0.0f;
            out[(long)(mbase + r) * SOUT + m] = val;
        }
    }
}

// deg starts at 1.0 (self-loop contribution)
__global__ void gcn_init_deg(float* __restrict__ deg, int N)
{
    int i = blockIdx.x * blockDim.x + threadIdx.x;
    if (i < N) deg[i] = 1.0f;
}

__global__ void gcn_count_deg(const int* __restrict__ col, float* __restrict__ deg, int E)
{
    int e = blockIdx.x * blockDim.x + threadIdx.x;
    if (e < E) atomicAdd(&deg[col[e]], 1.0f);
}

// deg >= 1 always (self-loops), so rsqrt is safe; store dinv in place
__global__ void gcn_make_dinv(float* __restrict__ deg, int N)
{
    int i = blockIdx.x * blockDim.x + threadIdx.x;
    if (i < N) deg[i] = rsqrtf(deg[i]);
}

// agg[i] = dinv[i]^2 * t[i]  (self-loop term; dense write initializes agg)
__global__ void gcn_self_init(const float4* __restrict__ t, const float* __restrict__ dinv,
                              float4* __restrict__ agg, int N)
{
    int i = blockIdx.x * blockDim.x + threadIdx.x;
    if (i < N) {
        float w = dinv[i] * dinv[i];
        float4 v = t[i];
        float4 r; r.x = w * v.x; r.y = w * v.y; r.z = w * v.z; r.w = w * v.w;
        agg[i] = r;
    }
}

// one thread per edge: coalesced index reads, one float4 gather (L2-resident),
// 4 f32 atomics into one cacheline of agg (L2-resident)
__global__ void gcn_edge_agg(const int* __restrict__ row, const int* __restrict__ col,
                             const float* __restrict__ dinv,
                             const float4* __restrict__ t,
                             float* __restrict__ agg, int E)
{
    int e = blockIdx.x * blockDim.x + threadIdx.x;
    if (e >= E) return;
    int r = row[e];
    int c = col[e];
    float w = dinv[r] * dinv[c];
    float4 v = t[r];
    float* dst = agg + (long)c * 4;
    atomicAdd(dst + 0, w * v.x);
    atomicAdd(dst + 1, w * v.y);
    atomicAdd(dst + 2, w * v.z);
    atomicAdd(dst + 3, w * v.w);
}

// one thread per node: h[i] = tanh(agg[i] + b), padded features forced to 0
__global__ void gcn_bias_tanh(const float4* __restrict__ agg, const float* __restrict__ b,
                              int Fout, float4* __restrict__ h, int N)
{
    int i = blockIdx.x * blockDim.x + threadIdx.x;
    if (i >= N) return;
    float4 v = agg[i];
    float4 r;
    r.x = tanhf(v.x + b[0]);
    r.y = tanhf(v.y + b[1]);
    r.z = (Fout > 2) ? tanhf(v.z + b[2]) : 0.0f;
    r.w = (Fout > 3) ? tanhf(v.w + b[3]) : 0.0f;
    h[i] = r;
}

// strip stride-4 padding: write h3 [N,3] contiguous
__global__ void gcn_copy_h3(const float* __restrict__ h, float* __restrict__ out, int N3)
{
    int tid = blockIdx.x * blockDim.x + threadIdx.x;
    if (tid >= N3) return;
    int i = tid / 3, f = tid - 3 * i;
    out[tid] = h[(long)i * 4 + f];
}

extern "C" void kernel_launch(void* const* d_in, const int* in_sizes, int n_in,
                              void* d_out, int out_size, void* d_ws, size_t ws_size,
                              hipStream_t stream)
{
    const float* x  = (const float*)d_in[0];
    const int*   ei = (const int*)d_in[1];
    const float* W1 = (const float*)d_in[2];
    const float* b1 = (const float*)d_in[3];
    const float* W2 = (const float*)d_in[4];
    const float* b2 = (const float*)d_in[5];
    const float* W3 = (const float*)d_in[6];
    const float* b3 = (const float*)d_in[7];
    const float* Wc = (const float*)d_in[8];
    const float* bc = (const float*)d_in[9];
    float* out = (float*)d_out;

    const int N = in_sizes[0] / 3;
    const int E = in_sizes[1] / 2;
    const int* row = ei;       // sources
    const int* col = ei + E;   // targets

    float* dinv = (float*)d_ws;                 // N
    float* t    = dinv + N;                     // 4N  (transformed features)
    float* agg  = t + (size_t)4 * N;            // 4N  (aggregation accumulator)
    float* h    = agg + (size_t)4 * N;          // 4N  (activations)

    const int B = 256;
    const int gN  = (N + B - 1) / B;
    const int gE  = (E + B - 1) / B;
    const int gN3 = (3 * N + B - 1) / B;
    const int gW  = (N + 127) / 128;            // 8 waves/block * 16 nodes/wave

    // degree + normalization
    gcn_init_deg<<<gN, B, 0, stream>>>(dinv, N);
    gcn_count_deg<<<gE, B, 0, stream>>>(col, dinv, E);
    gcn_make_dinv<<<gN, B, 0, stream>>>(dinv, N);

    // layer 1: x[N,3] -> h[N,4]
    gcn_wmma_transform<3, 3, 4, 4, 4, false><<<gW, B, 0, stream>>>(x, W1, nullptr, t, N);
    gcn_self_init<<<gN, B, 0, stream>>>((const float4*)t, dinv, (float4*)agg, N);
    gcn_edge_agg<<<gE, B, 0, stream>>>(row, col, dinv, (const float4*)t, agg, E);
    gcn_bias_tanh<<<gN, B, 0, stream>>>((const float4*)agg, b1, 4, (float4*)h, N);

    // layer 2: h[N,4] -> h[N,4]
    gcn_wmma_transform<4, 4, 4, 4, 4, false><<<gW, B, 0, stream>>>(h, W2, nullptr, t, N);
    gcn_self_init<<<gN, B, 0, stream>>>((const float4*)t, dinv, (float4*)agg, N);
    gcn_edge_agg<<<gE, B, 0, stream>>>(row, col, dinv, (const float4*)t, agg, E);
    gcn_bias_tanh<<<gN, B, 0, stream>>>((const float4*)agg, b2, 4, (float4*)h, N);

    // layer 3: h[N,4] -> h[N,3] (padded to 4)
    gcn_wmma_transform<4, 4, 3, 4, 4, false><<<gW, B, 0, stream>>>(h, W3, nullptr, t, N);
    gcn_self_init<<<gN, B, 0, stream>>>((const float4*)t, dinv, (float4*)agg, N);
    gcn_edge_agg<<<gE, B, 0, stream>>>(row, col, dinv, (const float4*)t, agg, E);
    gcn_bias_tanh<<<gN, B, 0, stream>>>((const float4*)agg, b3, 3, (float4*)h, N);

    // classifier: out[N,5] = h3 @ Wc + bc  (direct into d_out, stride 5)
    gcn_wmma_transform<4, 3, 5, 5, 0, true><<<gW, B, 0, stream>>>(h, Wc, bc, out, N);

    // second output: h3 [N,3] contiguous after out
    gcn_copy_h3<<<gN3, B, 0, stream>>>(h, out + (size_t)5 * N, 3 * N);
}